// CoreAttention_20246475833907
// MI455X (gfx1250) — compile-verified
//
#include <hip/hip_runtime.h>
#include <math.h>

typedef float v2f __attribute__((ext_vector_type(2)));
typedef float v8f __attribute__((ext_vector_type(8)));

#define GA_HEADS    8
#define GA_HEAD_DIM 16
#define GA_HIDDEN   128
#define GA_TILE     16

// ---------------------------------------------------------------------------
// Kernel 1: zero the accumulators (wV lives in d_out, Z in d_ws).
// ---------------------------------------------------------------------------
__global__ void ga_zero_kernel(float* __restrict__ wV, float* __restrict__ Z,
                               long long nO, long long nZ) {
    long long i      = (long long)blockIdx.x * blockDim.x + threadIdx.x;
    long long stride = (long long)gridDim.x * blockDim.x;
    for (long long j = i; j < nO; j += stride) wV[j] = 0.0f;
    for (long long j = i; j < nZ; j += stride) Z[j]  = 0.0f;
}

// ---------------------------------------------------------------------------
// Kernel 2: one wave32 per tile of 16 edges.
//   Scores via V_WMMA_F32_16X16X4_F32 diagonal trick (exact f32 math),
//   then exp(clip(./4)) and f32-atomic scatter of v[src]*score into wV, Z.
// ---------------------------------------------------------------------------
__global__ __launch_bounds__(32) void ga_edge_kernel(
    const float*     __restrict__ q,
    const float*     __restrict__ k,
    const float*     __restrict__ v,
    const long long* __restrict__ ei,   // [2, E] int64
    float*           __restrict__ wV,   // [N, 128] accumulator
    float*           __restrict__ Z,    // [N, 8]  accumulator
    int E)
{
    __shared__ float sc[GA_TILE * GA_HEADS];  // raw dots, then exp'd scores
    __shared__ int   sSrc[GA_TILE];
    __shared__ int   sDst[GA_TILE];

    const int lane = threadIdx.x;      // 0..31
    const int m    = lane & 15;        // edge column within tile
    const int hi   = lane >> 4;        // lane-group (selects K pair / row half)
    const long long e0 = (long long)blockIdx.x * GA_TILE;

    // --- per-column edge endpoints (clamped for tail tiles) -----------------
    long long eIdx  = e0 + m;
    long long eClmp = (eIdx < E) ? eIdx : (long long)(E - 1);
    int src = (int)ei[eClmp];                 // row 0 of edge_index gathers k,v
    int dst = (int)ei[(long long)E + eClmp];  // row 1 gathers q, scatter target

    if (hi == 0) { sSrc[m] = src; sDst[m] = dst; }

    const float* qd = q + (long long)dst * GA_HIDDEN;
    const float* ks = k + (long long)src * GA_HIDDEN;

    // speculative prefetch of next tile's edge indices (global_prefetch_b8)
    {
        long long nx = e0 + GA_TILE + m;
        if (nx >= E) nx = E - 1;
        __builtin_prefetch(&ei[nx], 0, 1);
        __builtin_prefetch(&ei[(long long)E + nx], 0, 1);
    }

    // --- 8 heads x 4 chained K=4 WMMAs; diag(D) = per-edge head dots --------
    // A(16x4): lanes 0-15 hold K=0,1 ; lanes 16-31 hold K=2,3  (ISA 7.12.2)
    // B(4x16): lanes 0-15 rows 0,1 ; lanes 16-31 rows 2,3 ; col = lane&15
    // Both reduce to: this lane loads dims {base, base+1}, base = 4*kk + 2*hi,
    // A from q[dst_m], B from k[src_m].
    v8f acc[GA_HEADS] = {};
    #pragma unroll
    for (int h = 0; h < GA_HEADS; ++h) {
        #pragma unroll
        for (int kk = 0; kk < 4; ++kk) {
            const int base = h * GA_HEAD_DIM + kk * 4 + 2 * hi;
            v2f a = *(const v2f*)(qd + base);
            v2f b = *(const v2f*)(ks + base);
            acc[h] = __builtin_amdgcn_wmma_f32_16x16x4_f32(
                false, a, false, b, (short)0, acc[h], false, false);
        }
    }

    // --- diagonal extraction: D[e][e] sits in acc[h][e&7] of lane
    //     (e<8 ? e : 16+e).  Select element with an unrolled cndmask chain. --
    {
        const int  idx    = m & 7;
        const bool isDiag = (hi == (m >> 3));
        if (isDiag) {
            #pragma unroll
            for (int h = 0; h < GA_HEADS; ++h) {
                float d = 0.0f;
                #pragma unroll
                for (int r = 0; r < 8; ++r)
                    if (r == idx) d = acc[h][r];
                sc[m * GA_HEADS + h] = d;
            }
        }
    }
    __syncthreads();

    // --- score transform: exp(clip(dot/sqrt(16), -5, 5)); invalid edges -> 0
    #pragma unroll
    for (int i = 0; i < 4; ++i) {
        const int   flat = lane * 4 + i;     // [0,128): e = flat>>3, h = flat&7
        const int   e    = flat >> 3;
        float raw = sc[flat] * 0.25f;
        raw = fminf(fmaxf(raw, -5.0f), 5.0f);
        float s = __expf(raw);
        if (e0 + e >= E) s = 0.0f;
        sc[flat] = s;
    }
    __syncthreads();

    // --- scatter: lane owns dims [4*lane, 4*lane+4) => head = lane>>2 -------
    const int h_of = lane >> 2;
    const int doff = lane * 4;               // == h_of*16 + (lane&3)*4

    for (int e = 0; e < GA_TILE; ++e) {
        if (e0 + e >= E) break;              // uniform
        const float s  = sc[e * GA_HEADS + h_of];
        const int   se = sSrc[e];
        const int   de = sDst[e];
        const float4 vv = *(const float4*)(v + (long long)se * GA_HIDDEN + doff);
        float* outp = wV + (long long)de * GA_HIDDEN + doff;
        unsafeAtomicAdd(outp + 0, vv.x * s);   // native global_atomic_add_f32
        unsafeAtomicAdd(outp + 1, vv.y * s);
        unsafeAtomicAdd(outp + 2, vv.z * s);
        unsafeAtomicAdd(outp + 3, vv.w * s);
        if ((lane & 3) == 0)
            unsafeAtomicAdd(&Z[(long long)de * GA_HEADS + h_of], s);
    }
}

// ---------------------------------------------------------------------------
// Kernel 3: x = wV / (Z + 1e-6), in place on d_out.
// ---------------------------------------------------------------------------
__global__ void ga_norm_kernel(float* __restrict__ wV, const float* __restrict__ Z,
                               long long nO) {
    long long i      = (long long)blockIdx.x * blockDim.x + threadIdx.x;
    long long stride = (long long)gridDim.x * blockDim.x;
    for (long long j = i; j < nO; j += stride) {
        const long long node = j >> 7;          // /128
        const int       h    = (int)((j >> 4) & 7);
        wV[j] = wV[j] / (Z[node * GA_HEADS + h] + 1e-6f);
    }
}

// ---------------------------------------------------------------------------
extern "C" void kernel_launch(void* const* d_in, const int* in_sizes, int n_in,
                              void* d_out, int out_size, void* d_ws, size_t ws_size,
                              hipStream_t stream) {
    const float*     q  = (const float*)d_in[0];
    const float*     k  = (const float*)d_in[1];
    const float*     v  = (const float*)d_in[2];
    const long long* ei = (const long long*)d_in[3];

    float* wV = (float*)d_out;
    float* Z  = (float*)d_ws;   // N*8 floats = 1.6 MB scratch

    const int N = in_sizes[0] / GA_HIDDEN;
    const int E = in_sizes[3] / 2;

    const long long nO = (long long)N * GA_HIDDEN;
    const long long nZ = (long long)N * GA_HEADS;

    ga_zero_kernel<<<2048, 256, 0, stream>>>(wV, Z, nO, nZ);

    const int tiles = (E + GA_TILE - 1) / GA_TILE;
    ga_edge_kernel<<<tiles, 32, 0, stream>>>(q, k, v, ei, wV, Z, E);

    const int threads = 256;
    const int blocks  = (int)((nO + threads - 1) / threads);
    ga_norm_kernel<<<blocks, threads, 0, stream>>>(wV, Z, nO);
}